// viola_attx_ddcm_dyk_29661044146255
// MI455X (gfx1250) — compile-verified
//
#include <hip/hip_runtime.h>
#include <math.h>

// ---------------- problem constants ----------------
#define BB 2
#define CC 64
#define HH 64   // also W and D
#define PLANE 4096           // 64*64
#define VOL   262144         // 64*64*64 per (b,c)

// workspace layout (float offsets)
#define PX   0        // pool_x  (B,C,64)
#define PY   8192     // pool_y
#define PZ   16384    // pool_z
#define VOFF 24576    // v[a][b][c][l]  3*8192
#define SOFF 49152    // sigmoid tables xs,ys,zs   3*8192
#define TOFF 73728    // t tables xt,yt,zt         3*8192
#define SSUM 98304    // per-batch sum of viola^2  (2 floats)
#define WS_FLOATS 98306

typedef __attribute__((ext_vector_type(2))) float v2f;
typedef __attribute__((ext_vector_type(4))) float v4f;
typedef __attribute__((ext_vector_type(8))) float v8f;

__device__ __forceinline__ float sigm(float v)  { return 1.0f / (1.0f + __expf(-v)); }
__device__ __forceinline__ float silu(float v)  { return v * sigm(v); }

__device__ __forceinline__ float viola_val(float Xs, float Xt, float Ys, float Zs,
                                           float Yt, float Zt) {
    float rel = Xt + Yt + Zt;
    rel = rel > 0.0f ? rel : 0.0f;
    float vv = Xs * Ys + Ys * Zs + Zs * Xs + Xs * Ys * Zs + rel;
    return 0.1f * vv + 0.3f;
}

// ---------------- kernel 0: zero accumulators ----------------
__global__ void k_zero(float* ws) {
    int i = blockIdx.x * 256 + threadIdx.x;
    if (i < WS_FLOATS) ws[i] = 0.0f;
}

// ---------------- kernel 1: triple axis pooling ----------------
// grid = B*C*8 blocks (8 h-chunks), 256 threads. Register-accumulated, LDS bins.
__global__ void k_pool(const float* __restrict__ x, float* __restrict__ ws) {
    int bid = blockIdx.x;
    int b  = bid >> 9;
    int c  = (bid >> 3) & 63;
    int hc = bid & 7;
    int tid = threadIdx.x;

    __shared__ float lx[8], ly[64], lz[64];
    if (tid < 8)  lx[tid] = 0.0f;
    if (tid < 64) { ly[tid] = 0.0f; lz[tid] = 0.0f; }
    __syncthreads();

    int dv = tid & 15;    // float4 index along d
    int wq = tid >> 4;    // base w
    const float* xb = x + (size_t)((b * 64 + c) * 64 + hc * 8) * PLANE;

    float ry[4] = {0,0,0,0};
    float rz[4] = {0,0,0,0};
    for (int hl = 0; hl < 8; ++hl) {
        const float* hb = xb + hl * PLANE;
        float sx = 0.0f;
        #pragma unroll
        for (int j = 0; j < 4; ++j) {
            int w = wq + 16 * j;
            v4f f = *reinterpret_cast<const v4f*>(hb + w * 64 + dv * 4);
            float s = f.x + f.y + f.z + f.w;
            sx += s; ry[j] += s;
            rz[0] += f.x; rz[1] += f.y; rz[2] += f.z; rz[3] += f.w;
        }
        atomicAdd(&lx[hl], sx);
    }
    #pragma unroll
    for (int j = 0; j < 4; ++j) atomicAdd(&ly[wq + 16 * j], ry[j]);
    #pragma unroll
    for (int j = 0; j < 4; ++j) atomicAdd(&lz[dv * 4 + j], rz[j]);
    __syncthreads();

    const float inv = 1.0f / 4096.0f;
    int row = (b * 64 + c) * 64;
    if (tid < 8)  ws[PX + row + hc * 8 + tid] = lx[tid] * inv;      // exclusive
    if (tid < 64) {
        atomicAdd(&ws[PY + row + tid], ly[tid] * inv);
        atomicAdd(&ws[PZ + row + tid], lz[tid] * inv);
    }
}

// ---------------- kernel 2: ch_norm + DDCM -> v[a][b][c][l] ----------------
// grid = 3 axes * B * 4 L-chunks = 24 blocks, 256 threads. All in LDS.
__global__ void k_ddcm(const float* __restrict__ ws_in, float* __restrict__ ws_out,
                       const float* __restrict__ gamma,
                       const float* __restrict__ w0, const float* __restrict__ w1,
                       const float* __restrict__ w2, const float* __restrict__ w3,
                       const float* __restrict__ w_out) {
    int bid = blockIdx.x;
    int a  = bid / (BB * 4);
    int rem = bid % (BB * 4);
    int b  = rem / 4;
    int l0 = (rem % 4) * 16;
    int tid = threadIdx.x;

    __shared__ float chan[9][64][16];   // slot0 = v; 1,2=y4; 3,4=y3; 5,6=y2; 7,8=y1
    __shared__ float ytmp[2][32][16];
    __shared__ float pv[64][16];
    __shared__ float mstat[16], istat[16];

    const float* pool = ws_in + a * 8192 + (size_t)(b * 64) * 64;
    for (int idx = tid; idx < 64 * 16; idx += 256) {
        int c = idx >> 4, l = idx & 15;
        pv[c][l] = pool[c * 64 + l0 + l];
    }
    __syncthreads();

    // channel norm across C per position l
    if (tid < 16) {
        float s = 0.0f, s2 = 0.0f;
        for (int c = 0; c < 64; ++c) { float v = pv[c][tid]; s += v; s2 += v * v; }
        float m = s * (1.0f / 64.0f);
        float var = s2 * (1.0f / 64.0f) - m * m;
        mstat[tid] = m;
        istat[tid] = rsqrtf(var + 1e-5f);
    }
    __syncthreads();
    for (int idx = tid; idx < 64 * 16; idx += 256) {
        int c = idx >> 4, l = idx & 15;
        chan[0][c][l] = (pv[c][l] - mstat[l]) * istat[l] * gamma[a * 64 + c];
    }
    __syncthreads();

    const int rates[4]   = {1, 7, 13, 19};
    const int strides[4] = {2, 2, 4, 4};
    const float* wptr[4] = {w0, w1, w2, w3};

    for (int n = 1; n <= 4; ++n) {
        int r = rates[n - 1], s = strides[n - 1];
        int Hout = 63 / s + 1;          // 32,32,16,16
        int cin = 2 * n - 1;
        int s0 = 9 - 2 * n;
        const float* W = wptr[n - 1];

        // dilated conv along C-dim (pointwise in l)
        int tot = 2 * Hout * 16;
        for (int idx = tid; idx < tot; idx += 256) {
            int l = idx & 15;
            int rest = idx >> 4;
            int hp = rest % Hout;
            int o  = rest / Hout;
            float acc = 0.0f;
            for (int ci = 0; ci < cin; ++ci) {
                int slot = (ci == cin - 1) ? 0 : (s0 + 2 + ci);
                const float* wk = W + ((a * 2 + o) * cin + ci) * 7;
                #pragma unroll
                for (int k = 0; k < 7; ++k) {
                    int row = hp * s + (k - 3) * r;
                    if (row >= 0 && row < 64) acc += wk[k] * chan[slot][row][l];
                }
            }
            ytmp[o][hp][l] = acc;
        }
        __syncthreads();

        // bilinear resize Hout -> 64 (half-pixel centers, edge clamp)
        float sc = (float)Hout * (1.0f / 64.0f);
        for (int idx = tid; idx < 2 * 64 * 16; idx += 256) {
            int l = idx & 15;
            int rest = idx >> 4;
            int i = rest & 63;
            int o = rest >> 6;
            float coord = ((float)i + 0.5f) * sc - 0.5f;
            int i0 = (int)floorf(coord);
            float t = coord - (float)i0;
            int ia = min(max(i0, 0), Hout - 1);
            int ib = min(max(i0 + 1, 0), Hout - 1);
            chan[s0 + o][i][l] = (1.0f - t) * ytmp[o][ia][l] + t * ytmp[o][ib][l];
        }
        __syncthreads();
    }

    // silu + 1x1 conv over 10 channels (ch 9 == ch 0 == v)
    for (int idx = tid; idx < 64 * 16; idx += 256) {
        int c = idx >> 4, l = idx & 15;
        float acc = 0.0f;
        #pragma unroll
        for (int ch = 0; ch < 10; ++ch) {
            float val = chan[ch == 9 ? 0 : ch][c][l];
            acc += w_out[a * 10 + ch] * silu(val);
        }
        ws_out[VOFF + a * 8192 + b * 4096 + c * 64 + l0 + l] = acc;
    }
}

// ---------------- kernel 3: sigmoid + groupnorm + tanh -> s/t tables ----------------
// grid = B blocks, 256 threads
__global__ void k_gate(float* __restrict__ ws, const float* __restrict__ gn_w,
                       const float* __restrict__ gn_b) {
    int b = blockIdx.x;
    int tid = threadIdx.x;
    __shared__ float lv[3 * 4096];
    __shared__ float red[256], red2[256];
    __shared__ float gm[2], gi[2];

    for (int idx = tid; idx < 3 * 4096; idx += 256) {
        int a = idx >> 12, r = idx & 4095;
        lv[idx] = ws[VOFF + a * 8192 + b * 4096 + r];
    }
    __syncthreads();

    // group norm stats: group g = channels [32g,32g+32), over concat L=192
    for (int g = 0; g < 2; ++g) {
        float s = 0.0f, s2 = 0.0f;
        for (int idx = tid; idx < 6144; idx += 256) {
            int c = g * 32 + idx / 192;
            int pos = idx % 192;
            int a = pos >> 6, l = pos & 63;
            float v = lv[a * 4096 + c * 64 + l];
            s += v; s2 += v * v;
        }
        red[tid] = s; red2[tid] = s2;
        __syncthreads();
        for (int st = 128; st > 0; st >>= 1) {
            if (tid < st) { red[tid] += red[tid + st]; red2[tid] += red2[tid + st]; }
            __syncthreads();
        }
        if (tid == 0) {
            float m = red[0] * (1.0f / 6144.0f);
            float var = red2[0] * (1.0f / 6144.0f) - m * m;
            gm[g] = m;
            gi[g] = rsqrtf(var + 1e-5f);
        }
        __syncthreads();
    }

    for (int idx = tid; idx < 3 * 4096; idx += 256) {
        int a = idx >> 12, r = idx & 4095;
        int c = r >> 6;
        float v = lv[idx];
        float sg = sigm(v);
        int g = c >> 5;
        float gn = (v - gm[g]) * gi[g] * gn_w[c] + gn_b[c];
        float tv = 0.5f * (tanhf(gn) + sg);
        ws[SOFF + a * 8192 + b * 4096 + r] = sg;
        ws[TOFF + a * 8192 + b * 4096 + r] = tv;
    }
}

// ---------------- kernel 4: sum of viola^2 via WMMA ones-trick ----------------
// grid = B*C*H = 8192 blocks, 256 threads. No x reads — pure table compute.
__global__ void k_norm(float* __restrict__ ws) {
    int bid = blockIdx.x;
    int b = bid >> 12;
    int c = (bid >> 6) & 63;
    int h = bid & 63;
    int tid = threadIdx.x;

    __shared__ float lys[64], lzs[64], lyt[64], lzt[64];
    __shared__ float red[256];
    int row = b * 4096 + c * 64;
    if (tid < 64) {
        lys[tid] = ws[SOFF + 8192  + row + tid];
        lzs[tid] = ws[SOFF + 16384 + row + tid];
        lyt[tid] = ws[TOFF + 8192  + row + tid];
        lzt[tid] = ws[TOFF + 16384 + row + tid];
    }
    float Xs = ws[SOFF + row + h];
    float Xt = ws[TOFF + row + h];
    __syncthreads();

    // Each wave feeds 64 fp32 values/iter through V_WMMA_F32_16X16X4_F32 against
    // an all-ones B: total = sum(all D entries)/16, layout-invariant.
    int lane = tid & 31;
    int wave = tid >> 5;
    v8f acc = {0.f, 0.f, 0.f, 0.f, 0.f, 0.f, 0.f, 0.f};
    v2f ones; ones.x = 1.0f; ones.y = 1.0f;
    for (int i = 0; i < 8; ++i) {
        int p = wave * 512 + i * 64 + lane * 2;
        v2f aa;
        {
            int w = p >> 6, d = p & 63;
            float pv = viola_val(Xs, Xt, lys[w], lzs[d], lyt[w], lzt[d]);
            aa.x = pv * pv;
        }
        {
            int p1 = p + 1;
            int w = p1 >> 6, d = p1 & 63;
            float pv = viola_val(Xs, Xt, lys[w], lzs[d], lyt[w], lzt[d]);
            aa.y = pv * pv;
        }
        acc = __builtin_amdgcn_wmma_f32_16x16x4_f32(false, aa, false, ones,
                                                    (short)0, acc, false, false);
    }
    float partial = acc[0] + acc[1] + acc[2] + acc[3] + acc[4] + acc[5] + acc[6] + acc[7];
    red[tid] = partial;
    __syncthreads();
    for (int st = 128; st > 0; st >>= 1) {
        if (tid < st) red[tid] += red[tid + st];
        __syncthreads();
    }
    if (tid == 0) atomicAdd(&ws[SSUM + b], red[0] * (1.0f / 16.0f));
}

// ---------------- kernel 5: out = x * viola * (1 + 1/nrm) ----------------
// grid = B*C*H = 8192 blocks, 256 threads; 128-bit streams, NT stores.
__global__ void k_apply(const float* __restrict__ x, const float* __restrict__ ws,
                        float* __restrict__ out) {
    int bid = blockIdx.x;
    int b = bid >> 12;
    int c = (bid >> 6) & 63;
    int h = bid & 63;
    int tid = threadIdx.x;

    __shared__ float lys[64], lzs[64], lyt[64], lzt[64];
    int row = b * 4096 + c * 64;
    if (tid < 64) {
        lys[tid] = ws[SOFF + 8192  + row + tid];
        lzs[tid] = ws[SOFF + 16384 + row + tid];
        lyt[tid] = ws[TOFF + 8192  + row + tid];
        lzt[tid] = ws[TOFF + 16384 + row + tid];
    }
    float Xs = ws[SOFF + row + h];
    float Xt = ws[TOFF + row + h];
    float nrm = sqrtf(ws[SSUM + b]);
    float scale = 1.0f + 1.0f / fmaxf(nrm, 1e-12f);
    __syncthreads();

    size_t base = (size_t)((b * 64 + c) * 64 + h) * PLANE;
    #pragma unroll
    for (int j = 0; j < 4; ++j) {
        int fidx = j * 256 + tid;        // 128-bit chunk index in [0,1024)
        int w = fidx >> 4;
        int dv = fidx & 15;
        v4f xv = *reinterpret_cast<const v4f*>(x + base + w * 64 + dv * 4);
        float Ys = lys[w], Yt = lyt[w];
        v4f ov;
        {
            int d = dv * 4;
            ov.x = xv.x * viola_val(Xs, Xt, Ys, lzs[d + 0], Yt, lzt[d + 0]) * scale;
            ov.y = xv.y * viola_val(Xs, Xt, Ys, lzs[d + 1], Yt, lzt[d + 1]) * scale;
            ov.z = xv.z * viola_val(Xs, Xt, Ys, lzs[d + 2], Yt, lzt[d + 2]) * scale;
            ov.w = xv.w * viola_val(Xs, Xt, Ys, lzs[d + 3], Yt, lzt[d + 3]) * scale;
        }
        __builtin_nontemporal_store(ov, reinterpret_cast<v4f*>(out + base + w * 64 + dv * 4));
    }
}

// ---------------- host ----------------
extern "C" void kernel_launch(void* const* d_in, const int* in_sizes, int n_in,
                              void* d_out, int out_size, void* d_ws, size_t ws_size,
                              hipStream_t stream) {
    const float* x     = (const float*)d_in[0];
    const float* gamma = (const float*)d_in[1];
    const float* w0    = (const float*)d_in[2];
    const float* w1    = (const float*)d_in[3];
    const float* w2    = (const float*)d_in[4];
    const float* w3    = (const float*)d_in[5];
    const float* w_out = (const float*)d_in[6];
    const float* gn_w  = (const float*)d_in[7];
    const float* gn_b  = (const float*)d_in[8];
    float* out = (float*)d_out;
    float* ws  = (float*)d_ws;

    k_zero <<<(WS_FLOATS + 255) / 256, 256, 0, stream>>>(ws);
    k_pool <<<BB * CC * 8, 256, 0, stream>>>(x, ws);
    k_ddcm <<<3 * BB * 4, 256, 0, stream>>>(ws, ws, gamma, w0, w1, w2, w3, w_out);
    k_gate <<<BB, 256, 0, stream>>>(ws, gn_w, gn_b);
    k_norm <<<BB * CC * HH, 256, 0, stream>>>(ws);
    k_apply<<<BB * CC * HH, 256, 0, stream>>>(x, ws, out);
}